// HymbaAttention_70574902608270
// MI455X (gfx1250) — compile-verified
//
#include <hip/hip_runtime.h>
#include <hip/hip_bf16.h>
#include <math.h>
#include <stdint.h>

// ---------------------------------------------------------------------------
// Hymba hybrid block for MI455X (gfx1250, wave32).
// - All GEMMs + attention matmuls on V_WMMA_F32_16X16X4_F32 (f32-exact).
// - GEMM tiles staged memory->LDS with GLOBAL_LOAD_ASYNC_TO_LDS_B128,
//   double-buffered and gated on ASYNCcnt (s_wait_asynccnt).
// - Fragment reads are 8B ds_load_b64 from loop-invariant bases, batched per
//   k-step so DS latency overlaps the WMMA pipe.
// - LDS strides chosen so the two lane-halves land on disjoint bank residues:
//   GEMM stride 36 (banks = 4r vs 4r+2 mod 64), attention stride 68 (same).
// ---------------------------------------------------------------------------

#define S_LEN   2048
#define DIM     1024
#define NH      16
#define KVH     4
#define HD      64
#define ROPE_D  32
#define INNER   2048
#define KCONV   4
#define DTR     64
#define NSTATE  16
#define KVDIM   (KVH * HD)              // 256
#define FUSED_N (2 * KVDIM + 2 * INNER) // 4608

typedef __attribute__((ext_vector_type(2))) float v2f;
typedef __attribute__((ext_vector_type(8))) float v8f;

__device__ __forceinline__ v8f wmma_f32_4(v2f a, v2f b, v8f c) {
  return __builtin_amdgcn_wmma_f32_16x16x4_f32(false, a, false, b, (short)0, c,
                                               false, false);
}

__device__ __forceinline__ float sigmoidf_(float x) { return 1.0f / (1.0f + __expf(-x)); }
__device__ __forceinline__ float siluf_(float x)    { return x * sigmoidf_(x); }
__device__ __forceinline__ float softplusf_(float x){ return (x > 20.0f) ? x : log1pf(__expf(x)); }

// Async memory->LDS copy of 16 bytes per lane. GVS addressing:
// mem = SGPR_base + VGPR_i32_offset; LDS dest = wave LDS base + VGPR offset.
__device__ __forceinline__ void async_b128(uint32_t lds_byte_off,
                                           const float* __restrict__ base,
                                           uint32_t glb_byte_off) {
  asm volatile("global_load_async_to_lds_b128 %0, %1, %2 offset:0"
               :: "v"(lds_byte_off), "v"(glb_byte_off), "s"(base)
               : "memory");
}
__device__ __forceinline__ void wait_async_le8() {
  asm volatile("s_wait_asynccnt 0x8" ::: "memory");
}
__device__ __forceinline__ void wait_async_0() {
  asm volatile("s_wait_asynccnt 0x0" ::: "memory");
}

// ---------------------------------------------------------------------------
// Generic NT GEMM: C[M,N] = A[M,K] * W[N,K]^T (row-dot-row).
// Block = 128 threads (4 waves), tile 64(M) x 64(N), BK = 32, double-buffered
// async LDS staging. LDS row stride 36 floats: 16B-aligned rows for B128,
// conflict-free fragment reads. Requires K % 32 == 0 (1024 / 2048 / 64).
// Dynamic LDS only => LDS offsets for async ops start at byte 0.
// ---------------------------------------------------------------------------
#define GEMM_LDSTRIDE 36
#define GEMM_TILE_F   (64 * GEMM_LDSTRIDE)              // floats per tile
#define GEMM_LDS_BYTES (4 * GEMM_TILE_F * 4)            // 2 buffers x (A+B)

__global__ __launch_bounds__(128) void gemm_nt_wmma(
    const float* __restrict__ A, const float* __restrict__ W, float* __restrict__ C,
    int M, int N, int K, int lda, int ldb, int ldc)
{
  extern __shared__ float lds[];   // [buf0: A,B][buf1: A,B], each 64x36 floats
  const int tid  = threadIdx.x;
  const int w    = tid >> 5;
  const int lane = tid & 31;
  const int lm   = lane & 15;
  const int lh   = lane >> 4;
  const int m0   = blockIdx.y * 64;
  const int n0   = blockIdx.x * 64;

  // Issue one K-chunk's A+B tiles as async B128 loads into buffer `buf`.
  // 64 rows x 32 floats = 512 x 16B chunks per tile; 128 threads x 4 rounds.
  auto issue_chunk = [&](int buf, int k0) {
    const uint32_t abase = (uint32_t)(buf * 2 * GEMM_TILE_F) * 4u;
    const uint32_t bbase = abase + (uint32_t)GEMM_TILE_F * 4u;
#pragma unroll
    for (int j = 0; j < 4; ++j) {
      int id = j * 128 + tid;            // [0,512)
      int r  = id >> 3;                  // row in tile
      int c4 = (id & 7) * 4;             // float col (16B chunk)
      int gm = m0 + r; if (gm > M - 1) gm = M - 1;   // clamp: junk rows unused
      async_b128(abase + (uint32_t)(r * GEMM_LDSTRIDE + c4) * 4u,
                 A, (uint32_t)((size_t)gm * lda + k0 + c4) * 4u);
    }
#pragma unroll
    for (int j = 0; j < 4; ++j) {
      int id = j * 128 + tid;
      int r  = id >> 3;
      int c4 = (id & 7) * 4;
      int gn = n0 + r; if (gn > N - 1) gn = N - 1;
      async_b128(bbase + (uint32_t)(r * GEMM_LDSTRIDE + c4) * 4u,
                 W, (uint32_t)((size_t)gn * ldb + k0 + c4) * 4u);
    }
  };

  v8f acc[4] = {};

  issue_chunk(0, 0);
  for (int k0 = 0; k0 < K; k0 += 32) {
    const int buf   = (k0 >> 5) & 1;
    const bool more = (k0 + 32) < K;
    if (more) issue_chunk(buf ^ 1, k0 + 32);     // prefetch next chunk
    if (more) wait_async_le8();                   // oldest 8 (this buf) done
    else      wait_async_0();
    __syncthreads();                              // tile visible to all waves

    const float* as = lds + buf * 2 * GEMM_TILE_F;
    const float* bs = as + GEMM_TILE_F;
    // Loop-invariant v2f bases: fragment (kk) = base[kk >> 1] (kk step 4 -> idx 2j).
    const v2f* ap  = (const v2f*)&as[(w * 16 + lm) * GEMM_LDSTRIDE + lh * 2];
    const v2f* bp0 = (const v2f*)&bs[( 0 + lm) * GEMM_LDSTRIDE + lh * 2];
    const v2f* bp1 = (const v2f*)&bs[(16 + lm) * GEMM_LDSTRIDE + lh * 2];
    const v2f* bp2 = (const v2f*)&bs[(32 + lm) * GEMM_LDSTRIDE + lh * 2];
    const v2f* bp3 = (const v2f*)&bs[(48 + lm) * GEMM_LDSTRIDE + lh * 2];
#pragma unroll
    for (int j = 0; j < 8; ++j) {                 // kk = 4*j
      v2f a = ap[2 * j];
      v2f bfrag[4];
      bfrag[0] = bp0[2 * j];
      bfrag[1] = bp1[2 * j];
      bfrag[2] = bp2[2 * j];
      bfrag[3] = bp3[2 * j];
#pragma unroll
      for (int t = 0; t < 4; ++t)
        acc[t] = wmma_f32_4(a, bfrag[t], acc[t]);
    }
    __syncthreads();  // all reads of this buffer done before it is re-filled
  }

  // C/D layout: lane holds column n = lm (+16t), element i -> row i + 8*lh.
#pragma unroll
  for (int t = 0; t < 4; ++t) {
    int gn = n0 + t * 16 + lm;
    if (gn < N) {
#pragma unroll
      for (int i = 0; i < 8; ++i) {
        int gm = m0 + w * 16 + 8 * lh + i;
        if (gm < M) C[(size_t)gm * ldc + gn] = acc[t][i];
      }
    }
  }
}

// ---------------------------------------------------------------------------
// QKV preprocessing: RMS-norm per head row, rotary on first 32 dims,
// q_gain and 1/sqrt(HD) folded into q. Emits head-major (h, s, d) layouts.
// ---------------------------------------------------------------------------
__device__ void norm_rope_row(const float* __restrict__ src, float* __restrict__ dst,
                              int s, float gain)
{
  float ss = 0.0f;
#pragma unroll
  for (int d = 0; d < HD; ++d) { float v = src[d]; ss += v * v; }
  float r = rsqrtf(ss * (1.0f / HD) + 1.1920929e-7f) * gain;
#pragma unroll
  for (int j = 0; j < ROPE_D / 2; ++j) {
    float invf = __powf(10000.0f, (-2.0f / ROPE_D) * (float)j);
    float ang = (float)s * invf;
    float sn, cs;
    __sincosf(ang, &sn, &cs);
    float x1 = src[j], x2 = src[ROPE_D / 2 + j];
    dst[j]              = (x1 * cs + x2 * sn) * r;
    dst[ROPE_D / 2 + j] = (-x1 * sn + x2 * cs) * r;
  }
#pragma unroll
  for (int d = ROPE_D; d < HD; ++d) dst[d] = src[d] * r;
}

__global__ __launch_bounds__(256) void qkv_preproc(
    const float* __restrict__ q_out, const float* __restrict__ fused,
    const float* __restrict__ q_gain,
    float* __restrict__ qn, float* __restrict__ kn, float* __restrict__ vn)
{
  int rid = blockIdx.x * blockDim.x + threadIdx.x;
  const int ROWS = NH + 2 * KVH; // 24 rows per s
  if (rid >= S_LEN * ROWS) return;
  int s = rid / ROWS;
  int t = rid % ROWS;
  if (t < NH) {
    int h = t;
    const float* src = &q_out[(size_t)s * DIM + h * HD];
    float* dst = &qn[((size_t)h * S_LEN + s) * HD];
    norm_rope_row(src, dst, s, q_gain[h] * 0.125f); // fold 1/sqrt(64)
  } else if (t < NH + KVH) {
    int kv = t - NH;
    const float* src = &fused[(size_t)s * FUSED_N + kv * HD];
    float* dst = &kn[((size_t)kv * S_LEN + s) * HD];
    norm_rope_row(src, dst, s, 1.0f);
  } else {
    int kv = t - NH - KVH;
    const float* src = &fused[(size_t)s * FUSED_N + KVDIM + kv * HD];
    float* dst = &vn[((size_t)kv * S_LEN + s) * HD];
#pragma unroll
    for (int d = 0; d < HD; ++d) dst[d] = src[d];
  }
}

// ---------------------------------------------------------------------------
// Causal flash attention. Block = (head, 64-query-row block), 128 threads.
// S=Q*K^T and O+=P*V both via WMMA f32 16x16x4. V reuses the K LDS buffer.
// LDS row stride 68: lane-halves hit disjoint bank residues (0 vs 2 mod 4),
// rows stay 8B/16B aligned for v2f fragment loads. ~53 KB LDS.
// ---------------------------------------------------------------------------
#define ATT_STRIDE 68

__global__ __launch_bounds__(128) void attn_fwd(
    const float* __restrict__ qn, const float* __restrict__ kn,
    const float* __restrict__ vn, float* __restrict__ attn_out)
{
  __shared__ float Qs[64 * ATT_STRIDE];
  __shared__ float KVs[64 * ATT_STRIDE];
  __shared__ float Ss[64 * ATT_STRIDE];
  __shared__ float m_row[64], l_row[64], scale_row[64];
  __shared__ float pmax[128], psum[128];

  const int tid = threadIdx.x;
  const int w = tid >> 5, lane = tid & 31, lm = lane & 15, lh = lane >> 4;
  const int qb = blockIdx.x;
  const int h  = blockIdx.y;
  const int kv = h >> 2;            // H/KVH = 4 (jnp.repeat -> h//4)
  const int q0 = qb * 64;

  for (int i = tid; i < 64 * 64; i += 128) {
    int r = i >> 6, d = i & 63;
    Qs[r * ATT_STRIDE + d] = qn[((size_t)h * S_LEN + q0 + r) * HD + d];
  }
  if (tid < 64) { m_row[tid] = -3.0e38f; l_row[tid] = 0.0f; }
  __syncthreads();

  v8f o[4] = {};

  const v2f* qp  = (const v2f*)&Qs[(w * 16 + lm) * ATT_STRIDE + lh * 2];
  const v2f* pp  = (const v2f*)&Ss[(w * 16 + lm) * ATT_STRIDE + lh * 2];
  const v2f* kp0 = (const v2f*)&KVs[( 0 + lm) * ATT_STRIDE + lh * 2];
  const v2f* kp1 = (const v2f*)&KVs[(16 + lm) * ATT_STRIDE + lh * 2];
  const v2f* kp2 = (const v2f*)&KVs[(32 + lm) * ATT_STRIDE + lh * 2];
  const v2f* kp3 = (const v2f*)&KVs[(48 + lm) * ATT_STRIDE + lh * 2];

  for (int kb = 0; kb <= qb; ++kb) {
    const int kstart = kb * 64;
    // --- K tile -> LDS (and prefetch next block's K/V rows) ---
    for (int i = tid; i < 64 * 64; i += 128) {
      int r = i >> 6, d = i & 63;
      KVs[r * ATT_STRIDE + d] = kn[((size_t)kv * S_LEN + kstart + r) * HD + d];
    }
    if (kb < qb) {
      int r = tid >> 1;  // 64 rows, 2 threads/row (global_prefetch_b8)
      __builtin_prefetch(&kn[((size_t)kv * S_LEN + kstart + 64 + r) * HD], 0, 0);
      __builtin_prefetch(&vn[((size_t)kv * S_LEN + kstart + 64 + r) * HD], 0, 0);
    }
    __syncthreads();

    // --- S = Q * K^T (scores), 16 k-steps over d=64, loads batched ---
    v8f sacc[4] = {};
#pragma unroll
    for (int j = 0; j < 16; ++j) {               // kk = 4*j
      v2f a = qp[2 * j];
      v2f bfrag[4];
      bfrag[0] = kp0[2 * j];
      bfrag[1] = kp1[2 * j];
      bfrag[2] = kp2[2 * j];
      bfrag[3] = kp3[2 * j];
#pragma unroll
      for (int t = 0; t < 4; ++t)
        sacc[t] = wmma_f32_4(a, bfrag[t], sacc[t]);
    }
#pragma unroll
    for (int t = 0; t < 4; ++t)
#pragma unroll
      for (int i = 0; i < 8; ++i)
        Ss[(w * 16 + 8 * lh + i) * ATT_STRIDE + t * 16 + lm] = sacc[t][i];
    __syncthreads();

    // --- online softmax: 2 threads per query row, 32 keys each ---
    const int r    = tid >> 1;
    const int half = tid & 1;
    const int qg   = q0 + r;
    float lmax = -3.0e38f;
    for (int k = half * 32; k < half * 32 + 32; ++k) {
      int kg = kstart + k;
      float sv = (kg <= qg) ? Ss[r * ATT_STRIDE + k] : -3.0e38f;
      lmax = fmaxf(lmax, sv);
    }
    pmax[tid] = lmax;
    __syncthreads();
    if (half == 0) {
      float mb = fmaxf(pmax[tid], pmax[tid + 1]);
      float mnew = fmaxf(m_row[r], mb);
      float sc = __expf(m_row[r] - mnew);
      m_row[r] = mnew;
      scale_row[r] = sc;
      l_row[r] *= sc;
    }
    __syncthreads();
    const float mnew = m_row[r];
    float ls = 0.0f;
    for (int k = half * 32; k < half * 32 + 32; ++k) {
      int kg = kstart + k;
      float p = (kg <= qg) ? __expf(Ss[r * ATT_STRIDE + k] - mnew) : 0.0f;
      Ss[r * ATT_STRIDE + k] = p;   // S tile becomes P tile
      ls += p;
    }
    psum[tid] = ls;
    __syncthreads();
    if (half == 0) l_row[r] += psum[tid] + psum[tid + 1];

    // --- V tile reuses KVs (K reads completed before Ss barrier) ---
    for (int i = tid; i < 64 * 64; i += 128) {
      int rr = i >> 6, d = i & 63;
      KVs[rr * ATT_STRIDE + d] = vn[((size_t)kv * S_LEN + kstart + rr) * HD + d];
    }
    __syncthreads();

    // --- rescale O, then O += P * V ---
    {
      const int rbase = w * 16 + 8 * lh;
      float sc[8];
#pragma unroll
      for (int i = 0; i < 8; ++i) sc[i] = scale_row[rbase + i];
#pragma unroll
      for (int t = 0; t < 4; ++t)
#pragma unroll
        for (int i = 0; i < 8; ++i) o[t][i] *= sc[i];
    }
#pragma unroll
    for (int j = 0; j < 16; ++j) {               // kk = 4*j over keys
      v2f a = pp[2 * j];
      v2f bfrag[4];
#pragma unroll
      for (int t = 0; t < 4; ++t) {
        // V is indexed [key][d]: b.x row kk+2lh, b.y row kk+2lh+1 (not contiguous)
        bfrag[t].x = KVs[(4 * j + lh * 2) * ATT_STRIDE + t * 16 + lm];
        bfrag[t].y = KVs[(4 * j + lh * 2 + 1) * ATT_STRIDE + t * 16 + lm];
      }
#pragma unroll
      for (int t = 0; t < 4; ++t)
        o[t] = wmma_f32_4(a, bfrag[t], o[t]);
    }
    __syncthreads(); // protect KVs/Ss for next key block
  }

  // --- normalize and store: attn_out[s][h*64+d] ---
  {
    const int rbase = w * 16 + 8 * lh;
    float inv[8];
#pragma unroll
    for (int i = 0; i < 8; ++i) inv[i] = 1.0f / l_row[rbase + i];
#pragma unroll
    for (int t = 0; t < 4; ++t) {
      int d = t * 16 + lm;
#pragma unroll
      for (int i = 0; i < 8; ++i) {
        int srow = q0 + rbase + i;
        attn_out[(size_t)srow * DIM + h * HD + d] = o[t][i] * inv[i];
      }
    }
  }
}

// ---------------------------------------------------------------------------
// Depthwise causal conv (K=4) + bias + SiLU. xs[s][c], coalesced over c.
// ---------------------------------------------------------------------------
__global__ __launch_bounds__(256) void conv_silu(
    const float* __restrict__ fused, const float* __restrict__ conv_w,
    const float* __restrict__ conv_b, float* __restrict__ xs)
{
  int id = blockIdx.x * 256 + threadIdx.x;
  if (id >= S_LEN * INNER) return;
  int c = id & (INNER - 1);
  int s = id >> 11;
  float acc = conv_b[c];
#pragma unroll
  for (int j = 0; j < KCONV; ++j) {
    int sp = s - (KCONV - 1) + j;
    if (sp >= 0)
      acc += fused[(size_t)sp * FUSED_N + 2 * KVDIM + c] * conv_w[c * KCONV + j];
  }
  xs[(size_t)s * INNER + c] = siluf_(acc);
}

// ---------------------------------------------------------------------------
// Selective scan: one thread per channel d (2048 total). State h[16] in
// registers; B/C rows (shared across channels) staged through LDS in
// 64-step chunks. Epilogue fuses D-skip and SiLU(gate).
// ---------------------------------------------------------------------------
__global__ __launch_bounds__(256) void ssm_scan(
    const float* __restrict__ dtb, const float* __restrict__ xs,
    const float* __restrict__ ssmp, const float* __restrict__ fused,
    const float* __restrict__ b_dt, const float* __restrict__ A_log,
    const float* __restrict__ D_param, float* __restrict__ scano)
{
  __shared__ float BC[64 * 32]; // 64 steps x (16 B + 16 C)
  const int d = blockIdx.x * 256 + threadIdx.x;

  float An[NSTATE];
#pragma unroll
  for (int n = 0; n < NSTATE; ++n) An[n] = -__expf(A_log[(size_t)d * NSTATE + n]);
  const float bdt = b_dt[d];
  const float Dp  = D_param[d];
  float hst[NSTATE];
#pragma unroll
  for (int n = 0; n < NSTATE; ++n) hst[n] = 0.0f;

  for (int s0 = 0; s0 < S_LEN; s0 += 64) {
    __syncthreads();
    for (int i = threadIdx.x; i < 64 * 32; i += 256) {
      int st = i >> 5, c = i & 31;
      BC[i] = ssmp[(size_t)(s0 + st) * 96 + DTR + c];
    }
    if (s0 + 64 < S_LEN) {
      // prefetch next chunk's B/C rows (global_prefetch_b8)
      int st = threadIdx.x & 63;
      __builtin_prefetch(&ssmp[(size_t)(s0 + 64 + st) * 96 + DTR], 0, 0);
    }
    __syncthreads();
    for (int st = 0; st < 64; ++st) {
      int s = s0 + st;
      float delta = softplusf_(dtb[(size_t)s * INNER + d] + bdt);
      float u = xs[(size_t)s * INNER + d];
      float du = delta * u;
      const float* Bv = &BC[st * 32];
      const float* Cv = &BC[st * 32 + 16];
      float y = 0.0f;
#pragma unroll
      for (int n = 0; n < NSTATE; ++n) {
        float dA = __expf(delta * An[n]);
        hst[n] = dA * hst[n] + du * Bv[n];
        y += hst[n] * Cv[n];
      }
      float g = fused[(size_t)s * FUSED_N + 2 * KVDIM + INNER + d];
      scano[(size_t)s * INNER + d] = (y + u * Dp) * siluf_(g);
    }
  }
}

// ---------------------------------------------------------------------------
// merged = sigmoid(alpha)*attn + (1-sigmoid(alpha))*mamba
// ---------------------------------------------------------------------------
__global__ __launch_bounds__(256) void merge_branches(
    const float* __restrict__ attn, const float* __restrict__ mamba,
    const float* __restrict__ merge_alpha, float* __restrict__ merged)
{
  int id = blockIdx.x * 256 + threadIdx.x;
  if (id >= S_LEN * DIM) return;
  float wgt = sigmoidf_(merge_alpha[0]);
  merged[id] = wgt * attn[id] + (1.0f - wgt) * mamba[id];
}

// ---------------------------------------------------------------------------
extern "C" void kernel_launch(void* const* d_in, const int* in_sizes, int n_in,
                              void* d_out, int out_size, void* d_ws, size_t ws_size,
                              hipStream_t stream)
{
  const float* x           = (const float*)d_in[0];
  const float* W_cq        = (const float*)d_in[1];
  const float* W_in        = (const float*)d_in[2];
  const float* q_gain      = (const float*)d_in[3];
  const float* conv_w      = (const float*)d_in[4];
  const float* conv_b      = (const float*)d_in[5];
  const float* W_xproj     = (const float*)d_in[6];
  const float* W_dt        = (const float*)d_in[7];
  const float* b_dt        = (const float*)d_in[8];
  const float* A_log       = (const float*)d_in[9];
  const float* D_param     = (const float*)d_in[10];
  const float* W_mout      = (const float*)d_in[11];
  const float* W_proj      = (const float*)d_in[12];
  const float* merge_alpha = (const float*)d_in[13];
  float* out = (float*)d_out;

  // Workspace carving (floats), with reuse: mamba_out aliases qn (dead after
  // attention), merged aliases q_out (dead after preproc). Total ~115 MB.
  float* ws = (float*)d_ws;
  size_t off = 0;
  float* qout  = ws + off; off += (size_t)S_LEN * DIM;        // 2M
  float* fused = ws + off; off += (size_t)S_LEN * FUSED_N;    // 9.44M
  float* qn    = ws + off; off += (size_t)NH  * S_LEN * HD;   // 2M
  float* kn    = ws + off; off += (size_t)KVH * S_LEN * HD;   // 0.5M
  float* vn    = ws + off; off += (size_t)KVH * S_LEN * HD;   // 0.5M
  float* attn  = ws + off; off += (size_t)S_LEN * DIM;        // 2M
  float* xs    = ws + off; off += (size_t)S_LEN * INNER;      // 4M
  float* ssmp  = ws + off; off += (size_t)S_LEN * 96;         // 0.19M
  float* dtb   = ws + off; off += (size_t)S_LEN * INNER;      // 4M
  float* scano = ws + off; off += (size_t)S_LEN * INNER;      // 4M
  float* mamba  = qn;   // reuse
  float* merged = qout; // reuse
  (void)off; (void)ws_size; (void)in_sizes; (void)n_in; (void)out_size;

  const size_t gemm_lds = GEMM_LDS_BYTES; // 36,864 B dynamic LDS

  // 1) q_out = x @ W_cq^T            (2048 x 1024, K=1024)
  gemm_nt_wmma<<<dim3(DIM / 64, S_LEN / 64), 128, gemm_lds, stream>>>(
      x, W_cq, qout, S_LEN, DIM, DIM, DIM, DIM, DIM);
  // 2) fused = x @ W_in^T            (2048 x 4608, K=1024)
  gemm_nt_wmma<<<dim3(FUSED_N / 64, S_LEN / 64), 128, gemm_lds, stream>>>(
      x, W_in, fused, S_LEN, FUSED_N, DIM, DIM, DIM, FUSED_N);
  // 3) qkv preprocessing (norm + rope + gain folds)
  qkv_preproc<<<(S_LEN * (NH + 2 * KVH) + 255) / 256, 256, 0, stream>>>(
      qout, fused, q_gain, qn, kn, vn);
  // 4) causal flash attention
  attn_fwd<<<dim3(S_LEN / 64, NH), 128, 0, stream>>>(qn, kn, vn, attn);
  // 5) depthwise causal conv + SiLU
  conv_silu<<<(S_LEN * INNER + 255) / 256, 256, 0, stream>>>(fused, conv_w, conv_b, xs);
  // 6) ssm_params = xs @ W_xproj^T   (2048 x 96, K=2048)
  gemm_nt_wmma<<<dim3((96 + 63) / 64, S_LEN / 64), 128, gemm_lds, stream>>>(
      xs, W_xproj, ssmp, S_LEN, 96, INNER, INNER, INNER, 96);
  // 7) dt = dt_low @ W_dt^T          (2048 x 2048, K=64; dt_low = ssmp[:, :64] via lda=96)
  gemm_nt_wmma<<<dim3(INNER / 64, S_LEN / 64), 128, gemm_lds, stream>>>(
      ssmp, W_dt, dtb, S_LEN, INNER, DTR, 96, DTR, INNER);
  // 8) selective scan (fused D-skip + SiLU(gate))
  ssm_scan<<<INNER / 256, 256, 0, stream>>>(dtb, xs, ssmp, fused, b_dt, A_log,
                                            D_param, scano);
  // 9) mamba_out = scan_out @ W_mout^T  (2048 x 1024, K=2048)
  gemm_nt_wmma<<<dim3(DIM / 64, S_LEN / 64), 128, gemm_lds, stream>>>(
      scano, W_mout, mamba, S_LEN, DIM, INNER, INNER, INNER, DIM);
  // 10) merge branches
  merge_branches<<<(S_LEN * DIM + 255) / 256, 256, 0, stream>>>(
      attn, mamba, merge_alpha, merged);
  // 11) out = merged @ W_proj^T      (2048 x 1024, K=1024)
  gemm_nt_wmma<<<dim3(DIM / 64, S_LEN / 64), 128, gemm_lds, stream>>>(
      merged, W_proj, out, S_LEN, DIM, DIM, DIM, DIM, DIM);
}